// Model_71073118814376
// MI455X (gfx1250) — compile-verified
//
#include <hip/hip_runtime.h>
#include <hip/hip_bf16.h>

typedef __attribute__((ext_vector_type(2))) float v2f;
typedef __attribute__((ext_vector_type(8))) float v8f;

static constexpr int kB   = 32;    // batches
static constexpr int kL   = 1024;  // sequence length
static constexpr int kC   = 34;    // raw channels (we only use channel 2)
static constexpr int kD   = 256;   // hidden dim
static constexpr int kE   = 8;     // experts
static constexpr int kP   = 4;     // live patches (p*256 < 1024)
static constexpr int kPL  = 16;    // patch length
static constexpr int kPred= 96;    // prediction length
static constexpr int kT   = kB * kP; // 128 live (batch,patch) rows

// ---------------------------------------------------------------------------
// Kernel 1: per-batch normalization stats + patch projection + router softmax
// One block per batch b; operates on x[b, :, 2] (stride kC).
// ---------------------------------------------------------------------------
__global__ void k_norm_proj_gate(const float* __restrict__ x,
                                 const float* __restrict__ Wp,  // (256,16)
                                 const float* __restrict__ bp,  // (256)
                                 const float* __restrict__ Wr,  // (8,256)
                                 const float* __restrict__ br,  // (8)
                                 float* __restrict__ ws_xp,     // (128,256)
                                 float* __restrict__ ws_gates,  // (128,8)
                                 float* __restrict__ ws_mean,   // (32)
                                 float* __restrict__ ws_std)    // (32)
{
    __shared__ float2 red[256];
    __shared__ float  xc[kP * kPL];        // 64 normalized samples
    __shared__ float  xpT[kP * kD];        // 4x256 projected patches
    __shared__ float  logits[kP * kE];

    const int b   = blockIdx.x;
    const int tid = threadIdx.x;
    const float* xb = x + (size_t)b * kL * kC + 2;

    // mean / variance over 1024 strided samples
    float s1 = 0.f, s2 = 0.f;
    #pragma unroll
    for (int j = 0; j < 4; ++j) {
        float v = xb[(size_t)(tid + 256 * j) * kC];
        s1 += v; s2 += v * v;
    }
    red[tid] = make_float2(s1, s2);
    __syncthreads();
    for (int off = 128; off > 0; off >>= 1) {
        if (tid < off) {
            float2 a = red[tid], c = red[tid + off];
            red[tid] = make_float2(a.x + c.x, a.y + c.y);
        }
        __syncthreads();
    }
    const float mean = red[0].x * (1.0f / kL);
    const float var  = red[0].y * (1.0f / kL) - mean * mean;
    const float stdv = sqrtf(var + 1e-5f);
    const float inv  = 1.0f / stdv;
    if (tid == 0) { ws_mean[b] = mean; ws_std[b] = stdv; }

    if (tid < kP * kPL) xc[tid] = (xb[(size_t)tid * kC] - mean) * inv;
    __syncthreads();

    // patch projection: each thread owns output dim o = tid, all 4 patches
    const int o = tid;
    #pragma unroll
    for (int p = 0; p < kP; ++p) {
        float acc = bp[o];
        #pragma unroll
        for (int i = 0; i < kPL; ++i) acc += xc[p * kPL + i] * Wp[o * kPL + i];
        xpT[p * kD + o] = acc;
        ws_xp[((size_t)(b * kP + p)) * kD + o] = acc;
    }
    __syncthreads();

    // router logits
    if (tid < kP * kE) {
        const int p = tid >> 3, e = tid & 7;
        float acc = br[e];
        for (int oo = 0; oo < kD; ++oo) acc += xpT[p * kD + oo] * Wr[e * kD + oo];
        logits[tid] = acc;
    }
    __syncthreads();

    // softmax over 8 experts, one thread per patch
    if (tid < kP) {
        const int p = tid;
        float mx = -3.0e38f;
        #pragma unroll
        for (int e = 0; e < kE; ++e) mx = fmaxf(mx, logits[p * kE + e]);
        float ex[kE], sum = 0.f;
        #pragma unroll
        for (int e = 0; e < kE; ++e) { ex[e] = expf(logits[p * kE + e] - mx); sum += ex[e]; }
        const float rs = 1.0f / sum;
        #pragma unroll
        for (int e = 0; e < kE; ++e)
            ws_gates[((size_t)(b * kP + p)) * kE + e] = ex[e] * rs;
    }
}

// ---------------------------------------------------------------------------
// Kernel 2: expert GEMM (128x256x256 per expert, 8 experts) via f32 WMMA,
// gate-combine + expert bias + de-normalization, scatter into flat(32,1024).
// One wave computes one 16x16 tile over all experts. EXEC all-1s (no div).
// ---------------------------------------------------------------------------
__global__ void k_experts_wmma(const float* __restrict__ xp,     // (128,256)
                               const float* __restrict__ gates,  // (128,8)
                               const float* __restrict__ We,     // (8,256,256) [e][h][d]
                               const float* __restrict__ be,     // (8,256)
                               const float* __restrict__ meanv,  // (32)
                               const float* __restrict__ stdv,   // (32)
                               float* __restrict__ flat)         // (32,1024)
{
    const int tile = blockIdx.x;      // 128 tiles: 8 (M) x 16 (N)
    const int tm   = tile >> 4;
    const int tn   = tile & 15;
    const int lane = threadIdx.x;
    const int half = lane >> 4;
    const int l15  = lane & 15;
    const int mrow = tm * 16 + l15;   // A row this lane supplies
    const int ncol = tn * 16 + l15;   // B/C column this lane owns

    v8f acc[kE];
    #pragma unroll
    for (int e = 0; e < kE; ++e)
        acc[e] = (v8f){0.f,0.f,0.f,0.f,0.f,0.f,0.f,0.f};

    const float* Arow = xp + (size_t)mrow * kD;
    for (int k = 0; k < kD; k += 4) {
        const int k0 = k + half * 2;                 // 8B aligned
        const v2f a = *(const v2f*)(Arow + k0);      // {A[m][k0], A[m][k0+1]}
        #pragma unroll
        for (int e = 0; e < kE; ++e) {
            const float* Wb = We + (size_t)e * kD * kD + (size_t)ncol * kD;
            const v2f bf = *(const v2f*)(Wb + k0);   // B[k][n] = We[e][n][k]
            acc[e] = __builtin_amdgcn_wmma_f32_16x16x4_f32(
                false, a, false, bf, (short)0, acc[e], false, false);
        }
    }

    // gate-weighted combine + bias, then denormalize and scatter
    float comb[8];
    #pragma unroll
    for (int r = 0; r < 8; ++r) comb[r] = 0.f;
    #pragma unroll
    for (int e = 0; e < kE; ++e) {
        const float bias = be[e * kD + ncol];
        #pragma unroll
        for (int r = 0; r < 8; ++r) {
            const int trow = tm * 16 + r + half * 8;     // C: VGPR r <-> M=r+8*half
            comb[r] += gates[trow * kE + e] * (acc[e][r] + bias);
        }
    }
    #pragma unroll
    for (int r = 0; r < 8; ++r) {
        const int trow = tm * 16 + r + half * 8;
        const int b = trow >> 2, p = trow & 3;
        flat[(size_t)b * kL + p * kD + ncol] = comb[r] * stdv[b] + meanv[b];
    }
}

// ---------------------------------------------------------------------------
// Kernel 3: head GEMM out(32,96) = flat(32,1024) @ W_head^T + b_head via WMMA.
// grid (2,6) tiles of 16x16; 256 f32 WMMA k-steps per tile.
// ---------------------------------------------------------------------------
__global__ void k_head_wmma(const float* __restrict__ flat,  // (32,1024)
                            const float* __restrict__ Wh,    // (96,1024)
                            const float* __restrict__ bh,    // (96)
                            float* __restrict__ out)         // (32,96)
{
    const int tm   = blockIdx.x;   // 0..1
    const int tn   = blockIdx.y;   // 0..5
    const int lane = threadIdx.x;
    const int half = lane >> 4;
    const int l15  = lane & 15;
    const int mrow = tm * 16 + l15;   // batch index for A fragment
    const int ncol = tn * 16 + l15;   // output j for B/C fragment

    v8f acc = (v8f){0.f,0.f,0.f,0.f,0.f,0.f,0.f,0.f};
    const float* Ar = flat + (size_t)mrow * kL;
    const float* Br = Wh   + (size_t)ncol * kL;   // B[k][n] = Wh[n][k]
    for (int k = 0; k < kL; k += 4) {
        const int k0 = k + half * 2;
        const v2f a  = *(const v2f*)(Ar + k0);
        const v2f bf = *(const v2f*)(Br + k0);
        acc = __builtin_amdgcn_wmma_f32_16x16x4_f32(
            false, a, false, bf, (short)0, acc, false, false);
    }
    const float bias = bh[ncol];
    #pragma unroll
    for (int r = 0; r < 8; ++r) {
        const int b = tm * 16 + r + half * 8;
        out[b * kPred + ncol] = acc[r] + bias;
    }
}

// ---------------------------------------------------------------------------
extern "C" void kernel_launch(void* const* d_in, const int* in_sizes, int n_in,
                              void* d_out, int out_size, void* d_ws, size_t ws_size,
                              hipStream_t stream) {
    const float* x  = (const float*)d_in[0];
    const float* Wp = (const float*)d_in[4];
    const float* bp = (const float*)d_in[5];
    const float* Wr = (const float*)d_in[6];
    const float* br = (const float*)d_in[7];
    const float* We = (const float*)d_in[8];
    const float* be = (const float*)d_in[9];
    const float* Wh = (const float*)d_in[10];
    const float* bh = (const float*)d_in[11];

    float* ws       = (float*)d_ws;
    float* ws_xp    = ws;                 // 128*256  = 32768 floats
    float* ws_gates = ws + 32768;         // 128*8    = 1024
    float* ws_mean  = ws + 33792;         // 32
    float* ws_std   = ws + 33824;         // 32
    float* ws_flat  = ws + 33856;         // 32*1024  = 32768

    k_norm_proj_gate<<<kB, 256, 0, stream>>>(x, Wp, bp, Wr, br,
                                             ws_xp, ws_gates, ws_mean, ws_std);
    k_experts_wmma<<<kT, 32, 0, stream>>>(ws_xp, ws_gates, We, be,
                                          ws_mean, ws_std, ws_flat);
    dim3 g3(2, 6);
    k_head_wmma<<<g3, 32, 0, stream>>>(ws_flat, Wh, bh, (float*)d_out);
}